// DifferentiableBezierRenderer_30597347016922
// MI455X (gfx1250) — compile-verified
//
#include <hip/hip_runtime.h>
#include <hip/hip_bf16.h>

// ---------------------------------------------------------------------------
// Problem constants (match the JAX reference)
// ---------------------------------------------------------------------------
#define IMG_H    512
#define IMG_W    512
#define NSAMP    32          // samples per Bezier segment
#define NSEG     64          // (193-1)/3 cubic segments
#define NEDGE    2048        // NSEG*NSAMP path points == edges (closed path)

typedef __attribute__((ext_vector_type(2))) float v2f;
typedef __attribute__((ext_vector_type(8))) float v8f;

// ---------------------------------------------------------------------------
// Kernel 1: Bezier path sampling as a true matmul on the WMMA unit.
//   pts(32x2 per segment) = T(32x4 Bernstein) x P(4x2 control points)
// Batched: A-block = 16 t-samples x 4 basis fns (f32 16x4),
//          B-block = 4 basis fns x 16 columns (8 segments x {x,y}),
//          D = 16x16 f32 -> 16 samples x 16 (seg,coord) columns.
// 2 row-blocks x 8 segment-blocks = 16 x V_WMMA_F32_16X16X4_F32 for the
// whole 2048-point path.  One wave32, EXEC all ones (WMMA requirement).
//
// VGPR layouts per ISA 7.12.2:
//   A 16x4 f32 : lane L holds row M=L%16; VGPR0/1 = K = 2*(L/16), 2*(L/16)+1
//   B 4x16 f32 : lane L holds col N=L%16; VGPR0/1 = K = 2*(L/16), 2*(L/16)+1
//   D 16x16 f32: lane L holds col N=L%16; VGPR r  = M = r + 8*(L/16)
// ---------------------------------------------------------------------------
__global__ __launch_bounds__(32)
void bezier_sample_wmma(const float* __restrict__ cp,   // 193 x 2 control pts
                        float* __restrict__ path)       // 2048 x 2 out
{
    const int lane  = threadIdx.x;       // 0..31
    const int kh    = lane >> 4;         // K-half: 0 -> K={0,1}, 1 -> K={2,3}
    const int l16   = lane & 15;         // row (A) / column (B,D) index
    const int segc  = l16 >> 1;          // segment within B-block (0..7)
    const int coord = l16 & 1;           // 0 = x, 1 = y

    for (int sb = 0; sb < 8; ++sb) {     // segment blocks: 8 segs each
        const int seg  = sb * 8 + segc;
        const int base = 3 * seg + 2 * kh;      // control-point row for K
        v2f B;
        B.x = cp[(base + 0) * 2 + coord];       // P[K  ][coord]
        B.y = cp[(base + 1) * 2 + coord];       // P[K+1][coord]

        #pragma unroll
        for (int rb = 0; rb < 2; ++rb) {        // sample row blocks (16 each)
            const int i  = rb * 16 + l16;       // sample index 0..31
            const float t  = (float)i * (1.0f / 31.0f);   // linspace(0,1,32)
            const float mt = 1.0f - t;
            const float w0 = mt * mt * mt;
            const float w1 = 3.0f * mt * mt * t;
            const float w2 = 3.0f * mt * t * t;
            const float w3 = t * t * t;
            v2f A;
            A.x = kh ? w2 : w0;                 // Bernstein basis, K and K+1
            A.y = kh ? w3 : w1;

            v8f C = {};                          // accumulate from zero
            C = __builtin_amdgcn_wmma_f32_16x16x4_f32(
                    /*neg_a=*/false, A, /*neg_b=*/false, B,
                    /*c_mod=*/(short)0, C, /*reuse_a=*/false, /*reuse_b=*/false);

            #pragma unroll
            for (int r = 0; r < 8; ++r) {
                const int M      = r + 8 * kh;          // sample row in block
                const int sample = rb * 16 + M;         // 0..31 within segment
                path[(seg * NSAMP + sample) * 2 + coord] = C[r];
            }
        }
    }
}

// ---------------------------------------------------------------------------
// Kernel 2: soft winding number, one block per image row.
// Phase 1: per-row per-edge constants (x_cross, w) -> LDS (ds_load_b64 later).
// Phase 2: each of 128 threads owns 4 consecutive pixels; per edge:
//          E0 = exp(gx - xc); E_{k} = E0 * e^k  (1 v_exp + 3 v_mul)
//          sigma = rcp(1 + E_k)                 (4 v_rcp)
//          wind += w * sigma                    (4 v_fma)
// exp overflow -> inf -> rcp(inf)=0 == saturated sigmoid (correct).
// ---------------------------------------------------------------------------
__global__ __launch_bounds__(128)
void winding_rows(const float* __restrict__ path,
                  const float* __restrict__ color,
                  float* __restrict__ out)
{
    __shared__ float2 s_e[NEDGE];        // (x_cross, w) per edge: 16 KB

    const int   gy  = blockIdx.x;
    const int   tid = threadIdx.x;
    const float fy  = (float)gy;

    // ---- Phase 1: per-row edge table (2048 edges / 128 threads = 16 each)
    for (int e = tid; e < NEDGE; e += 128) {
        const int   e1 = (e + 1) & (NEDGE - 1);
        const float x0 = path[e * 2 + 0];
        const float y0 = path[e * 2 + 1];
        const float x1 = path[e1 * 2 + 0];
        const float y1 = path[e1 * 2 + 1];
        const float dy  = y1 - y0;
        const float inv = __builtin_amdgcn_rcpf(dy + 1e-8f);
        const float t   = (fy - y0) * inv;
        const float xc  = x0 + t * (x1 - x0);
        const float v0  = 1.0f / (1.0f + __expf(-20.0f * t));
        const float v1  = 1.0f / (1.0f + __expf(-20.0f * (1.0f - t)));
        const float sgn = (dy > 0.0f) ? 1.0f : ((dy < 0.0f) ? -1.0f : 0.0f);
        const float msk = (fabsf(dy) >= 1e-6f) ? 1.0f : 0.0f;
        s_e[e] = make_float2(xc, v0 * v1 * sgn * msk);
    }
    __syncthreads();

    // ---- Phase 2: 4 consecutive pixels per thread
    const float gx0 = (float)(4 * tid);
    const float E1C = 2.7182818284f;             // e^1
    const float E2C = 7.3890560989f;             // e^2
    const float E3C = 20.0855369232f;            // e^3
    float w0 = 0.0f, w1 = 0.0f, w2 = 0.0f, w3 = 0.0f;

    #pragma unroll 8
    for (int e = 0; e < NEDGE; ++e) {
        const float2 ew = s_e[e];                // LDS broadcast, ds_load_b64
        const float  E0 = __expf(gx0 - ew.x);    // sigma(xc-gx)=1/(1+exp(gx-xc))
        const float  Ea = E0 * E1C;
        const float  Eb = E0 * E2C;
        const float  Ec = E0 * E3C;
        w0 += ew.y * __builtin_amdgcn_rcpf(1.0f + E0);
        w1 += ew.y * __builtin_amdgcn_rcpf(1.0f + Ea);
        w2 += ew.y * __builtin_amdgcn_rcpf(1.0f + Eb);
        w3 += ew.y * __builtin_amdgcn_rcpf(1.0f + Ec);
    }

    // ---- Phase 3: alpha = sigmoid(4*wind); write RGBA (float4, coalesced)
    const float r = color[0], g = color[1], b = color[2];
    const float a0 = 1.0f / (1.0f + __expf(-4.0f * w0));
    const float a1 = 1.0f / (1.0f + __expf(-4.0f * w1));
    const float a2 = 1.0f / (1.0f + __expf(-4.0f * w2));
    const float a3 = 1.0f / (1.0f + __expf(-4.0f * w3));

    float4* o4 = (float4*)out;
    const int px = gy * IMG_W + 4 * tid;
    o4[px + 0] = make_float4(r, g, b, a0);
    o4[px + 1] = make_float4(r, g, b, a1);
    o4[px + 2] = make_float4(r, g, b, a2);
    o4[px + 3] = make_float4(r, g, b, a3);
}

// ---------------------------------------------------------------------------
extern "C" void kernel_launch(void* const* d_in, const int* in_sizes, int n_in,
                              void* d_out, int out_size, void* d_ws, size_t ws_size,
                              hipStream_t stream)
{
    const float* cp    = (const float*)d_in[0];   // 193 x 2, fp32
    const float* color = (const float*)d_in[1];   // 3, fp32
    float*       out   = (float*)d_out;           // 512 x 512 x 4, fp32
    float*       path  = (float*)d_ws;            // 2048 x 2 fp32 scratch

    // 1) Path sampling via WMMA matmul (single wave32)
    bezier_sample_wmma<<<1, 32, 0, stream>>>(cp, path);

    // 2) One block per row; 128 threads x 4 pixels
    winding_rows<<<IMG_H, 128, 0, stream>>>(path, color, out);
}